// LinearAttention_37160057045028
// MI455X (gfx1250) — compile-verified
//
#include <hip/hip_runtime.h>
#include <math.h>

typedef float v2f __attribute__((ext_vector_type(2)));
typedef float v4f __attribute__((ext_vector_type(4)));
typedef float v8f __attribute__((ext_vector_type(8)));

#define L_DIM   4096
#define D_DIM   64
#define NCHUNK  8
#define CHUNK_L (L_DIM / NCHUNK)   // 512
#define SUB_L   64
#define ROWS_P2 128
#define QPAD    66                 // even pad: aligned b64 A-frags, conflict-safe
#define EPS_F   1e-6f

__device__ __forceinline__ float elu1(float x) {
    // elu(x) + 1  ==  x>0 ? x+1 : exp(x)
    return x > 0.0f ? x + 1.0f : __expf(x);
}

// K-pair-interleaved LDS layout: element (k, c) at [(k>>1)*128 + c*2 + (k&1)]
// => the (k, k+1) pair needed by a WMMA fragment is one aligned b64.

// -------- Phase 1: partial KV = (elu(K)+1)^T @ V  and partial Ksum, per (head, chunk)
__global__ __launch_bounds__(256)
void la_phase1(const float* __restrict__ Kin, const float* __restrict__ Vin,
               float* __restrict__ KVpart, float* __restrict__ KsumPart)
{
    __shared__ float Ks2[SUB_L * D_DIM];   // 16 KB, pair-interleaved
    __shared__ float Vs2[SUB_L * D_DIM];   // 16 KB, pair-interleaved
    __shared__ float red[256];

    const int head  = blockIdx.y;
    const int chunk = blockIdx.x;
    const int tid   = threadIdx.x;
    const int lane  = tid & 31;
    const int wave  = tid >> 5;
    const int lhalf = lane >> 4;          // 0 or 1
    const int lm    = lane & 15;

    const float* Kh = Kin + (size_t)head * L_DIM * D_DIM + (size_t)chunk * CHUNK_L * D_DIM;
    const float* Vh = Vin + (size_t)head * L_DIM * D_DIM + (size_t)chunk * CHUNK_L * D_DIM;

    // wave -> two 16x16 output tiles
    const int mt  = wave >> 1;
    const int nt0 = (wave & 1) * 2;

    v8f acc0 = {0.f,0.f,0.f,0.f,0.f,0.f,0.f,0.f};
    v8f acc1 = acc0;

    float ksum = 0.0f;
    const int kd = tid & 63;   // column for ksum
    const int kg = tid >> 6;   // row-group 0..3

    const v2f* K2 = (const v2f*)Ks2;
    const v2f* V2 = (const v2f*)Vs2;

    for (int sub = 0; sub < CHUNK_L / SUB_L; ++sub) {
        const v4f* K4 = (const v4f*)(Kh + sub * SUB_L * D_DIM);
        const v4f* V4 = (const v4f*)(Vh + sub * SUB_L * D_DIM);
        __syncthreads();
        #pragma unroll
        for (int i = 0; i < (SUB_L * D_DIM) / (256 * 4); ++i) {
            const int idx4 = tid + i * 256;          // 0..1023
            const int row  = idx4 >> 4;
            const int col  = (idx4 & 15) * 4;
            const v4f kv = K4[idx4];
            const v4f vv = V4[idx4];
            const int base = (row >> 1) * 128 + col * 2 + (row & 1);
            Ks2[base + 0] = elu1(kv.x);
            Ks2[base + 2] = elu1(kv.y);
            Ks2[base + 4] = elu1(kv.z);
            Ks2[base + 6] = elu1(kv.w);
            Vs2[base + 0] = vv.x;
            Vs2[base + 2] = vv.y;
            Vs2[base + 4] = vv.z;
            Vs2[base + 6] = vv.w;
        }
        __syncthreads();

        // ksum partial: 4 row-groups of 16 rows each, per column
        #pragma unroll
        for (int r = 0; r < 16; ++r) {
            const int row = kg * 16 + r;
            ksum += Ks2[(row >> 1) * 128 + kd * 2 + (row & 1)];
        }

        // K^T @ V over 64 local rows = 16 steps of K=4
        #pragma unroll
        for (int kk = 0; kk < 16; ++kk) {
            const int lp = kk * 2 + lhalf;           // k-pair index
            const v2f a  = K2[lp * 64 + mt * 16 + lm];        // A(M=d,K=l)=K'[l][d]
            const v2f b0 = V2[lp * 64 + nt0 * 16 + lm];       // B(K=l,N=v)=V[l][v]
            const v2f b1 = V2[lp * 64 + (nt0 + 1) * 16 + lm];
            acc0 = __builtin_amdgcn_wmma_f32_16x16x4_f32(false, a, false, b0,
                                                         (short)0, acc0, false, false);
            acc1 = __builtin_amdgcn_wmma_f32_16x16x4_f32(false, a, false, b1,
                                                         (short)0, acc1, false, false);
        }
    }

    // write exclusive partial KV slot (deterministic, no atomics)
    float* KVslot = KVpart + ((size_t)head * NCHUNK + chunk) * (D_DIM * D_DIM);
    #pragma unroll
    for (int r = 0; r < 8; ++r) {
        const int m = mt * 16 + r + lhalf * 8;                // D: M = r + 8*half
        KVslot[m * D_DIM + nt0 * 16 + lm]       = acc0[r];
        KVslot[m * D_DIM + (nt0 + 1) * 16 + lm] = acc1[r];
    }

    // reduce ksum row-groups
    red[tid] = ksum;
    __syncthreads();
    if (tid < 64) {
        float s = red[tid] + red[tid + 64] + red[tid + 128] + red[tid + 192];
        KsumPart[((size_t)head * NCHUNK + chunk) * D_DIM + tid] = s;
    }
}

// -------- Phase 2: out = (elu(Q)+1) @ KV, row-scaled by Z = 1/(Q'.Ksum + eps)
__global__ __launch_bounds__(256)
void la_phase2(const float* __restrict__ Qin, const float* __restrict__ KVpart,
               const float* __restrict__ KsumPart, float* __restrict__ Out)
{
    __shared__ float KVs2[D_DIM * D_DIM];         // 16 KB, pair-interleaved over k
    __shared__ float Ksum_s[D_DIM];
    __shared__ float Qs[ROWS_P2 * QPAD];          // ~33 KB, padded
    __shared__ float Zs[ROWS_P2];

    const int head  = blockIdx.y;
    const int rb    = blockIdx.x;                 // 128-row block, 0..31
    const int tid   = threadIdx.x;
    const int lane  = tid & 31;
    const int wave  = tid >> 5;
    const int lhalf = lane >> 4;
    const int lm    = lane & 15;

    const float* Qh = Qin + (size_t)head * L_DIM * D_DIM + (size_t)rb * ROWS_P2 * D_DIM;
    float*       Oh = Out + (size_t)head * L_DIM * D_DIM + (size_t)rb * ROWS_P2 * D_DIM;

    // reduce KV partials (L2-resident; deterministic order) into interleaved LDS
    const v4f* KVh4 = (const v4f*)(KVpart + (size_t)head * NCHUNK * D_DIM * D_DIM);
    for (int idx4 = tid; idx4 < (D_DIM * D_DIM) / 4; idx4 += 256) {
        v4f s = {0.f, 0.f, 0.f, 0.f};
        #pragma unroll
        for (int c = 0; c < NCHUNK; ++c) s += KVh4[c * (D_DIM * D_DIM / 4) + idx4];
        const int row  = idx4 >> 4;
        const int col  = (idx4 & 15) * 4;
        const int base = (row >> 1) * 128 + col * 2 + (row & 1);
        KVs2[base + 0] = s.x;
        KVs2[base + 2] = s.y;
        KVs2[base + 4] = s.z;
        KVs2[base + 6] = s.w;
    }
    if (tid < D_DIM) {
        const float* Ksh = KsumPart + (size_t)head * NCHUNK * D_DIM;
        float s = 0.f;
        #pragma unroll
        for (int c = 0; c < NCHUNK; ++c) s += Ksh[c * D_DIM + tid];
        Ksum_s[tid] = s;
    }
    // stage Q' (b128 global reads, padded LDS rows)
    const v4f* Q4 = (const v4f*)Qh;
    for (int idx4 = tid; idx4 < (ROWS_P2 * D_DIM) / 4; idx4 += 256) {
        const int row = idx4 >> 4;
        const int col = (idx4 & 15) * 4;
        const v4f qv  = Q4[idx4];
        v2f* dst = (v2f*)&Qs[row * QPAD + col];   // even offset -> aligned
        v2f lo = {elu1(qv.x), elu1(qv.y)};
        v2f hi = {elu1(qv.z), elu1(qv.w)};
        dst[0] = lo;
        dst[1] = hi;
    }
    __syncthreads();

    // Z per row
    if (tid < ROWS_P2) {
        float d = EPS_F;
        #pragma unroll
        for (int k = 0; k < D_DIM; ++k) d += Qs[tid * QPAD + k] * Ksum_s[k];
        Zs[tid] = 1.0f / d;
    }
    __syncthreads();

    // wave -> 16-row M tile, all four 16-col N tiles
    const int m0 = wave * 16;
    v8f acc0 = {0.f,0.f,0.f,0.f,0.f,0.f,0.f,0.f};
    v8f acc1 = acc0, acc2 = acc0, acc3 = acc0;

    const v2f* Q2  = (const v2f*)Qs;              // row stride = QPAD/2 = 33 v2f
    const v2f* KV2 = (const v2f*)KVs2;

    #pragma unroll
    for (int kk = 0; kk < 16; ++kk) {
        const int lp = kk * 2 + lhalf;            // k-pair index
        const v2f a  = Q2[(m0 + lm) * (QPAD / 2) + lp];       // A(M=l,K=d)=Q'[l][d]
        const v2f b0 = KV2[lp * 64 +  0 + lm];
        const v2f b1 = KV2[lp * 64 + 16 + lm];
        const v2f b2 = KV2[lp * 64 + 32 + lm];
        const v2f b3 = KV2[lp * 64 + 48 + lm];
        acc0 = __builtin_amdgcn_wmma_f32_16x16x4_f32(false, a, false, b0, (short)0, acc0, false, false);
        acc1 = __builtin_amdgcn_wmma_f32_16x16x4_f32(false, a, false, b1, (short)0, acc1, false, false);
        acc2 = __builtin_amdgcn_wmma_f32_16x16x4_f32(false, a, false, b2, (short)0, acc2, false, false);
        acc3 = __builtin_amdgcn_wmma_f32_16x16x4_f32(false, a, false, b3, (short)0, acc3, false, false);
    }

    #pragma unroll
    for (int r = 0; r < 8; ++r) {
        const int row = m0 + r + lhalf * 8;
        const float z = Zs[row];
        Oh[row * D_DIM +  0 + lm] = acc0[r] * z;
        Oh[row * D_DIM + 16 + lm] = acc1[r] * z;
        Oh[row * D_DIM + 32 + lm] = acc2[r] * z;
        Oh[row * D_DIM + 48 + lm] = acc3[r] * z;
    }
}

extern "C" void kernel_launch(void* const* d_in, const int* in_sizes, int n_in,
                              void* d_out, int out_size, void* d_ws, size_t ws_size,
                              hipStream_t stream) {
    const float* q = (const float*)d_in[0];
    const float* k = (const float*)d_in[1];
    const float* v = (const float*)d_in[2];
    float* out = (float*)d_out;

    const int heads = in_sizes[0] / (L_DIM * D_DIM);   // B*H = 32

    float* KVpart   = (float*)d_ws;                                      // heads*8*4096 floats
    float* KsumPart = KVpart + (size_t)heads * NCHUNK * D_DIM * D_DIM;   // heads*8*64 floats

    dim3 g1(NCHUNK, heads);
    la_phase1<<<g1, 256, 0, stream>>>(k, v, KVpart, KsumPart);

    dim3 g2(L_DIM / ROWS_P2, heads);
    la_phase2<<<g2, 256, 0, stream>>>(q, KVpart, KsumPart, out);
}